// CustomGAT_58884001628563
// MI455X (gfx1250) — compile-verified
//
#include <hip/hip_runtime.h>

typedef float v2f __attribute__((ext_vector_type(2)));
typedef float v8f __attribute__((ext_vector_type(8)));

#define C_DIM 64

// ---------------------------------------------------------------------------
// Stage A: p[i] = dot(x[i,:], w)  via V_WMMA_F32_16X16X4_F32.
// One wave32 computes 16 rows. A-frag (16x4 f32): lanes 0-15 hold K={0,1},
// lanes 16-31 hold K={2,3} (VGPR0/1). B-frag (4x16): every column set to w[k],
// so every column of D = the 16 row-dots; we read column 0 (lanes 0 and 16).
// ---------------------------------------------------------------------------
__global__ void __launch_bounds__(256)
gemv_dot_wmma(const float* __restrict__ x, const float* __restrict__ w,
              float* __restrict__ p, int n) {
  const int lane = threadIdx.x & 31;
  const int wave = blockIdx.x * (blockDim.x >> 5) + (threadIdx.x >> 5);
  const int row0 = wave << 4;                 // 16 rows per wave
  if (row0 >= n) return;                      // wave-uniform exit: EXEC stays all-1s
  const int m  = lane & 15;                   // row within tile
  const int kb = (lane >> 4) << 1;            // K base: 0 (lanes 0-15) or 2 (lanes 16-31)
  const float* xrow = x + (size_t)(row0 + m) * C_DIM + kb;
  v8f acc = {};
#pragma unroll
  for (int k0 = 0; k0 < C_DIM; k0 += 4) {
    v2f a; a.x = xrow[k0];     a.y = xrow[k0 + 1];
    v2f b; b.x = w[k0 + kb];   b.y = w[k0 + kb + 1];
    // (neg_a, A, neg_b, B, c_mod, C, reuse_a, reuse_b)
    acc = __builtin_amdgcn_wmma_f32_16x16x4_f32(false, a, false, b,
                                                (short)0, acc, false, false);
  }
  // D layout: VGPR g, lanes 0-15 -> M=g, N=lane; lanes 16-31 -> M=8+g, N=lane-16.
  if (m == 0) {                               // lanes 0 (rows 0-7) and 16 (rows 8-15)
    const int rbase = row0 + ((lane >> 4) << 3);
#pragma unroll
    for (int g = 0; g < 8; ++g) p[rbase + g] = acc[g];
  }
}

// ---------------------------------------------------------------------------
// Stage B: per edge e, learned[dst] += p[src] - p[dst] + lin_b
// (self-loop message == lin_b, folded into the score stage instead)
// ---------------------------------------------------------------------------
__global__ void __launch_bounds__(256)
edge_message_scatter(const int* __restrict__ src, const int* __restrict__ dst,
                     const float* __restrict__ p, const float* __restrict__ lin_b,
                     float* __restrict__ learned, int n) {
  int e = blockIdx.x * blockDim.x + threadIdx.x;
  if (e >= n) return;
  int s = src[e], d = dst[e];
  atomicAdd(&learned[d], p[s] - p[d] + lin_b[0]);
}

// ---------------------------------------------------------------------------
// Stage C+D fused: per edge e (note: score indexed by EDGE number, matching
// the reference's scores[:e] quirk), out[dst,:] += x[src,:] * score.
// One wave per edge, float2 per lane (64 channels), f32 global atomics.
// ---------------------------------------------------------------------------
__global__ void __launch_bounds__(256)
edge_aggregate(const float* __restrict__ x, const int* __restrict__ src,
               const int* __restrict__ dst, const float* __restrict__ learned,
               const float* __restrict__ lin_b, const float* __restrict__ w1,
               const float* __restrict__ w2, const float* __restrict__ w3,
               float* __restrict__ out, int n, int first) {
  long long t = (long long)blockIdx.x * blockDim.x + threadIdx.x;
  int e  = (int)(t >> 5);
  int c2 = (int)(t & 31);                     // covers channels 2*c2, 2*c2+1
  if (e >= n) return;
  int s = src[e], d = dst[e];
  float base  = (e < first) ? w1[0]
              : ((((e - first) & 1) == 0) ? w2[0] : w3[0]);
  float score = base + ((e >= first) ? (learned[e] + lin_b[0]) : 0.0f);
  const float2 xv = ((const float2*)(x + (size_t)s * C_DIM))[c2];
  float* o = out + (size_t)d * C_DIM + 2 * c2;
  atomicAdd(o,     xv.x * score);
  atomicAdd(o + 1, xv.y * score);
}

// ---------------------------------------------------------------------------
// Stage E: in-place ReLU on the aggregated output (float4 vectorized).
// ---------------------------------------------------------------------------
__global__ void __launch_bounds__(256)
relu_inplace(float* __restrict__ out, long long n4) {
  long long i = (long long)blockIdx.x * blockDim.x + threadIdx.x;
  if (i >= n4) return;
  float4 v = ((float4*)out)[i];
  v.x = fmaxf(v.x, 0.0f); v.y = fmaxf(v.y, 0.0f);
  v.z = fmaxf(v.z, 0.0f); v.w = fmaxf(v.w, 0.0f);
  ((float4*)out)[i] = v;
}

// ---------------------------------------------------------------------------
// Inputs (setup_inputs order): 0:x [N*64] f32, 1:lin_w [64] f32, 2:lin_b [1],
// 3:w1 [1], 4:w2 [1], 5:w3 [1], 6:edge_index [2*N] int (row0=src, row1=dst).
// Workspace: p (N f32) | learned (N f32)  -> 8 MB.
// ---------------------------------------------------------------------------
extern "C" void kernel_launch(void* const* d_in, const int* in_sizes, int n_in,
                              void* d_out, int out_size, void* d_ws, size_t ws_size,
                              hipStream_t stream) {
  const float* x     = (const float*)d_in[0];
  const float* lin_w = (const float*)d_in[1];
  const float* lin_b = (const float*)d_in[2];
  const float* w1    = (const float*)d_in[3];
  const float* w2    = (const float*)d_in[4];
  const float* w3    = (const float*)d_in[5];
  const int*   eidx  = (const int*)d_in[6];

  const int n     = in_sizes[6] / 2;          // num edges == num nodes
  const int first = n / 3;
  const int* src = eidx;
  const int* dst = eidx + n;

  float* p       = (float*)d_ws;
  float* learned = p + n;

  // Atomic accumulators must start from zero on EVERY call (graph replays).
  hipMemsetAsync(learned, 0, (size_t)n * sizeof(float), stream);
  hipMemsetAsync(d_out,   0, (size_t)out_size * sizeof(float), stream);

  { // Stage A: WMMA GEMV, 16 rows/wave, 8 waves/block
    int waves  = (n + 15) / 16;
    int blocks = (waves + 7) / 8;
    gemv_dot_wmma<<<blocks, 256, 0, stream>>>(x, lin_w, p, n);
  }
  { // Stage B: edge message scatter
    int blocks = (n + 255) / 256;
    edge_message_scatter<<<blocks, 256, 0, stream>>>(src, dst, p, lin_b, learned, n);
  }
  { // Stage C+D: gather-scale-scatter, one wave per edge
    long long threads = (long long)n * 32;
    int blocks = (int)((threads + 255) / 256);
    edge_aggregate<<<blocks, 256, 0, stream>>>(x, src, dst, learned,
                                               lin_b, w1, w2, w3,
                                               (float*)d_out, n, first);
  }
  { // Stage E: ReLU
    long long n4 = (long long)out_size / 4;
    int blocks = (int)((n4 + 255) / 256);
    relu_inplace<<<blocks, 256, 0, stream>>>((float*)d_out, n4);
  }
}